// TxaFilterComplexTorch_8091718385916
// MI455X (gfx1250) — compile-verified
//
#include <hip/hip_runtime.h>

// ---------------------------------------------------------------------------
// Grouped complex FIR (valid cross-correlation) via V_WMMA_F32_16X16X4_F32.
//
//  x: [B=32, S=16384, C=16, 2] f32 (re/im interleaved)
//  w: [C=16, K=256, 2]         f32
//  y: [B, L=16129, C, 1, 2]    f32
//
// Per (channel, batch-half): Y_tile(16 out-pos x 16 batches) = T @ X,
// T = Toeplitz of w (zero padded), complex folded into the real K axis:
//   A_r[m,2u+0]= wr[s-m]  A_r[m,2u+1]=-wi[s-m]   (accum Yr)
//   A_i[m,2u+0]= wi[s-m]  A_i[m,2u+1]= wr[s-m]   (accum Yi)
//   B  [2u+p,n]= x[n, l0+s_u, c, p]   <- contiguous b64 load (native layout)
// One ds_load_b64 (A) + one global_load_b64 (B, shared) feed 2 WMMAs.
// ---------------------------------------------------------------------------

typedef __attribute__((ext_vector_type(2))) float v2f;
typedef __attribute__((ext_vector_type(8))) float v8f;

#define N_CHANNELS 16
#define IN_SEQ     16384
#define OUT_SEQ    16129
#define CONV_SIZE  256
#define XPITCH     (N_CHANNELS * 2)          // floats per seq position

#define TILE_L          16
#define WAVES_PER_BLOCK 8
#define TILES_PER_CH    ((OUT_SEQ + TILE_L - 1) / TILE_L)                 // 1009
#define GROUPS          ((TILES_PER_CH + WAVES_PER_BLOCK - 1) / WAVES_PER_BLOCK) // 127

// Toeplitz window: s in [0, 271]; w index s-m in [-15, 271]; pad with zeros.
#define WIDX_PAD   15
#define W_ENTRIES  (WIDX_PAD + CONV_SIZE + TILE_L)   // 287 complex entries

__global__ __launch_bounds__(256) void txa_fir_wmma_f32(
    const float* __restrict__ x, const float* __restrict__ w,
    float* __restrict__ out)
{
    __shared__ v2f sAr[W_ENTRIES];   // {wr, -wi}, zero padded
    __shared__ v2f sAi[W_ENTRIES];   // {wi,  wr}, zero padded

    const int c   = blockIdx.z;      // channel
    const int h   = blockIdx.y;      // batch half (0: b 0-15, 1: b 16-31)
    const int tid = threadIdx.x;

    // Build padded Toeplitz weight tables for this channel.
    for (int e = tid; e < W_ENTRIES; e += 256) {
        const int iw = e - WIDX_PAD;
        v2f ar = {0.0f, 0.0f};
        v2f ai = {0.0f, 0.0f};
        if (iw >= 0 && iw < CONV_SIZE) {
            const float wr = w[(c * CONV_SIZE + iw) * 2 + 0];
            const float wi = w[(c * CONV_SIZE + iw) * 2 + 1];
            ar.x = wr;  ar.y = -wi;
            ai.x = wi;  ai.y =  wr;
        }
        sAr[e] = ar;
        sAi[e] = ai;
    }
    __syncthreads();

    const int wave = tid >> 5;
    const int lane = tid & 31;
    const int n    = lane & 15;      // batch column == Toeplitz row m
    const int hi   = lane >> 4;      // selects K-pair {2*hi, 2*hi+1}

    const int tile = blockIdx.x * WAVES_PER_BLOCK + wave;
    if (tile >= TILES_PER_CH) return;          // wave-uniform exit (after barrier)
    const int l0    = tile * TILE_L;
    const int batch = h * 16 + n;

    v8f Dr = {};   // Yr accumulator (16x16 f32, 8 VGPRs)
    v8f Di = {};   // Yi accumulator

    const float* __restrict__ xb =
        x + (size_t)batch * IN_SEQ * XPITCH + c * 2;
    int widx = hi - n + WIDX_PAD;    // LDS index; +2 per K-step

    if (l0 + 271 <= IN_SEQ - 1) {
        // Fast path: full window in range (all tiles except the last one).
        const float* __restrict__ px = xb + (size_t)(l0 + hi) * XPITCH;
#pragma unroll 4
        for (int S = 0; S < 272; S += 2) {
            const v2f b  = *(const v2f*)px;      // x{re,im} @ seq l0+S+hi
            const v2f ar = sAr[widx];
            const v2f ai = sAi[widx];
            Dr = __builtin_amdgcn_wmma_f32_16x16x4_f32(
                     false, ar, false, b, (short)0, Dr, false, false);
            Di = __builtin_amdgcn_wmma_f32_16x16x4_f32(
                     false, ai, false, b, (short)0, Di, false, false);
            px   += 2 * XPITCH;
            widx += 2;
        }
    } else {
        // Edge tile (l0 == 16128): clamp the sequence address. Clamped
        // garbage taps hit the zero-padded region of the weight tables
        // for every valid output row, so results stay exact.
        for (int S = 0; S < 272; S += 2) {
            int s = l0 + S + hi;
            s = (s > IN_SEQ - 1) ? (IN_SEQ - 1) : s;
            const v2f b  = *(const v2f*)(xb + (size_t)s * XPITCH);
            const v2f ar = sAr[widx];
            const v2f ai = sAi[widx];
            Dr = __builtin_amdgcn_wmma_f32_16x16x4_f32(
                     false, ar, false, b, (short)0, Dr, false, false);
            Di = __builtin_amdgcn_wmma_f32_16x16x4_f32(
                     false, ai, false, b, (short)0, Di, false, false);
            widx += 2;
        }
    }

    // Store: D element i of lane -> output row m = i + 8*hi, batch col n.
    // out[b, l, c, 0, {re,im}] as one b64 per element.
    v2f* __restrict__ o = (v2f*)out;
    const size_t obase = (size_t)batch * OUT_SEQ * N_CHANNELS + c;
#pragma unroll
    for (int i = 0; i < 8; ++i) {
        const int l = l0 + i + 8 * hi;
        if (l < OUT_SEQ) {
            v2f val;
            val.x = Dr[i];
            val.y = Di[i];
            o[obase + (size_t)l * N_CHANNELS] = val;
        }
    }
}

extern "C" void kernel_launch(void* const* d_in, const int* in_sizes, int n_in,
                              void* d_out, int out_size, void* d_ws, size_t ws_size,
                              hipStream_t stream) {
    const float* x = (const float*)d_in[0];   // [32,16384,16,2] f32
    const float* w = (const float*)d_in[1];   // [16,256,2] f32
    float* out = (float*)d_out;               // [32,16129,16,1,2] f32

    dim3 grid(GROUPS, 2, N_CHANNELS);         // 127 x 2 x 16 blocks
    txa_fir_wmma_f32<<<grid, 256, 0, stream>>>(x, w, out);
}